// WeatherLSTM_22368189677692
// MI455X (gfx1250) — compile-verified
//
#include <hip/hip_runtime.h>

typedef __attribute__((ext_vector_type(2))) float v2f;
typedef __attribute__((ext_vector_type(8))) float v8f;

#define LOG2E 1.4426950408889634f

#if __has_builtin(__builtin_amdgcn_tanhf)
// gfx1250 has a native V_TANH_F32 transcendental.
__device__ __forceinline__ float fast_tanh(float x) {
    return __builtin_amdgcn_tanhf(x);
}
__device__ __forceinline__ float fast_sigmoid(float x) {
    // sigmoid(x) = 0.5*tanh(x/2) + 0.5  -> mul + tanh + fma
    return __builtin_amdgcn_tanhf(0.5f * x) * 0.5f + 0.5f;
}
#else
__device__ __forceinline__ float fast_sigmoid(float x) {
    // 1 / (1 + exp(-x)) via v_exp_f32 + v_rcp_f32
    return __builtin_amdgcn_rcpf(1.0f + __builtin_amdgcn_exp2f(-LOG2E * x));
}
__device__ __forceinline__ float fast_tanh(float x) {
    // tanh(x) = 2*sigmoid(2x) - 1
    return 2.0f * __builtin_amdgcn_rcpf(1.0f + __builtin_amdgcn_exp2f(-2.0f * LOG2E * x)) - 1.0f;
}
#endif

// Combined weight matrix W_z[Kslot=0..11][gate-slot Mg=0..47].
//
// Gate-row permutation: lane-half h owns all 4 gates of hidden units 5h..5h+4:
//   slot s = 8*tile + (Mg%8), valid while s < 20,
//   unit = 5*(Mg%16 / 8) + s/4, gate-type g = s%4 (PyTorch order i,f,g,o).
//
// K-slot permutation: chosen so each lane-half's B-fragment K-slots are fed
// entirely by state that half already owns -> no cross-lane traffic in loop.
//   B layout (4x16 f32): lane l supplies K = p + 2*(l/16), p = VGPR 0/1,
//   i.e. half0 supplies slots {0,1,4,5,8,9}, half1 supplies {2,3,6,7,10,11}.
//   zmap[slot] -> logical z index: 0 = x_t, 1..10 = h[0..9], 11 = 1 (bias).
__constant__ int c_zmap[12] = {0, 1, 6, 7, 2, 3, 8, 9, 4, 5, 10, 11};

__device__ __forceinline__ float wz_elem(int Mg, int Kslot,
                                         const float* __restrict__ Wih,
                                         const float* __restrict__ Whh,
                                         const float* __restrict__ bih,
                                         const float* __restrict__ bhh) {
    int tile = Mg >> 4;
    int m    = Mg & 15;
    int h    = m >> 3;
    int r    = m & 7;
    int s    = tile * 8 + r;
    if (s >= 20) return 0.0f;                 // padding slot
    int unit = 5 * h + (s >> 2);
    int g    = s & 3;
    int row  = g * 10 + unit;                 // row in original [4H, .] matrices
    int K    = c_zmap[Kslot];
    if (K == 0)  return Wih[row];             // I == 1
    if (K <= 10) return Whh[row * 10 + (K - 1)];
    return bih[row] + bhh[row];               // K == 11: fused bias
}

__global__ __launch_bounds__(32)
void WeatherLSTM_wmma_kernel(const float* __restrict__ x,
                             const float* __restrict__ Wih,
                             const float* __restrict__ Whh,
                             const float* __restrict__ bih,
                             const float* __restrict__ bhh,
                             const float* __restrict__ Wfc,
                             const float* __restrict__ bfc,
                             float* __restrict__ out,
                             int T) {
    const int lane = threadIdx.x;   // 0..31, one wave per block
    const int n    = lane & 15;     // batch column within tile
    const int hi   = lane >> 4;     // lane half: owns hidden units 5*hi..5*hi+4
    const int row  = blockIdx.x * 16 + n;

    // ---- Preload A fragments (weights): 3 gate tiles x 3 K-chunks ----
    // A layout (16x4 f32): lane l: M = l%16; VGPR0 = Kslot 2*(l/16), VGPR1 = +1.
    v2f a[3][3];
#pragma unroll
    for (int t = 0; t < 3; ++t) {
#pragma unroll
        for (int kb = 0; kb < 3; ++kb) {
            int Mg = 16 * t + n;
            int K0 = 4 * kb + 2 * hi;
            a[t][kb].x = wz_elem(Mg, K0,     Wih, Whh, bih, bhh);
            a[t][kb].y = wz_elem(Mg, K0 + 1, Wih, Whh, bih, bhh);
        }
    }

    // Per-lane recurrent state: 5 hidden units each (unit = 5*hi + l)
    float hh[5] = {0.f, 0.f, 0.f, 0.f, 0.f};
    float cc[5] = {0.f, 0.f, 0.f, 0.f, 0.f};

    const float* xrow = x + (size_t)row * T;     // x[B,T,1], row-contiguous in t

    // Distance-2 prefetch: the buffer being consumed was loaded 2 outer
    // iterations (8 steps) ago, so the newest load can stay in flight.
    float4 buf0 = *(const float4*)xrow;                       // t 0..3
    float4 buf1 = make_float4(0.f, 0.f, 0.f, 0.f);
    if (T > 4) buf1 = *(const float4*)(xrow + 4);             // t 4..7

    for (int t4 = 0; t4 < T; t4 += 4) {
        float4 xnew = make_float4(0.f, 0.f, 0.f, 0.f);
        if (t4 + 8 < T) xnew = *(const float4*)(xrow + t4 + 8);
        float xs[4] = {buf0.x, buf0.y, buf0.z, buf0.w};

#pragma unroll
        for (int j = 0; j < 4; ++j) {
            float xt = xs[j];

            // B fragments: pure per-lane selects, no cross-lane ops.
            // half0 K-slot pairs {0,1}{4,5}{8,9} = (x,h0)(h1,h2)(h3,h4)
            // half1 K-slot pairs {2,3}{6,7}{10,11} = (h5,h6)(h7,h8)(h9,1)
            v2f b0, b1, b2;
            b0.x = hi ? hh[0] : xt;     b0.y = hi ? hh[1] : hh[0];
            b1.x = hi ? hh[2] : hh[1];  b1.y = hi ? hh[3] : hh[2];
            b2.x = hi ? hh[4] : hh[3];  b2.y = hi ? 1.0f  : hh[4];

            v8f z8 = {};
            v8f acc0 = __builtin_amdgcn_wmma_f32_16x16x4_f32(false, a[0][0], false, b0, (short)0, z8,   false, false);
            v8f acc1 = __builtin_amdgcn_wmma_f32_16x16x4_f32(false, a[1][0], false, b0, (short)0, z8,   false, false);
            v8f acc2 = __builtin_amdgcn_wmma_f32_16x16x4_f32(false, a[2][0], false, b0, (short)0, z8,   false, false);
            acc0     = __builtin_amdgcn_wmma_f32_16x16x4_f32(false, a[0][1], false, b1, (short)0, acc0, false, false);
            acc1     = __builtin_amdgcn_wmma_f32_16x16x4_f32(false, a[1][1], false, b1, (short)0, acc1, false, false);
            acc2     = __builtin_amdgcn_wmma_f32_16x16x4_f32(false, a[2][1], false, b1, (short)0, acc2, false, false);
            acc0     = __builtin_amdgcn_wmma_f32_16x16x4_f32(false, a[0][2], false, b2, (short)0, acc0, false, false);
            acc1     = __builtin_amdgcn_wmma_f32_16x16x4_f32(false, a[1][2], false, b2, (short)0, acc1, false, false);
            acc2     = __builtin_amdgcn_wmma_f32_16x16x4_f32(false, a[2][2], false, b2, (short)0, acc2, false, false);

            // Gate value for local unit l, gate g sits at flat s = 4l+g:
            // G[s] = acc[s/8][s%8]  (lane-half offset is implicit in D layout)
            float G[20];
#pragma unroll
            for (int s = 0; s < 8; ++s) G[s]      = acc0[s];
#pragma unroll
            for (int s = 0; s < 8; ++s) G[8 + s]  = acc1[s];
#pragma unroll
            for (int s = 0; s < 4; ++s) G[16 + s] = acc2[s];

#pragma unroll
            for (int l = 0; l < 5; ++l) {
                float ig = fast_sigmoid(G[4 * l + 0]);
                float fg = fast_sigmoid(G[4 * l + 1]);
                float gg = fast_tanh   (G[4 * l + 2]);
                float og = fast_sigmoid(G[4 * l + 3]);
                float c  = fg * cc[l] + ig * gg;
                cc[l] = c;
                hh[l] = og * fast_tanh(c);
            }
        }
        buf0 = buf1;
        buf1 = xnew;
    }

    // ---- Final FC on last hidden state: out[b] = h . W_fc + b_fc ----
    float partial = 0.f;
#pragma unroll
    for (int l = 0; l < 5; ++l) partial += hh[l] * Wfc[5 * hi + l];
    float total = partial + __shfl_xor(partial, 16, 32);  // full-EXEC shuffle
    if (hi == 0) out[row] = total + bfc[0];
}

extern "C" void kernel_launch(void* const* d_in, const int* in_sizes, int n_in,
                              void* d_out, int out_size, void* d_ws, size_t ws_size,
                              hipStream_t stream) {
    (void)n_in; (void)d_ws; (void)ws_size;
    const float* x_p   = (const float*)d_in[0];
    const float* Wih_p = (const float*)d_in[1];
    const float* Whh_p = (const float*)d_in[2];
    const float* bih_p = (const float*)d_in[3];
    const float* bhh_p = (const float*)d_in[4];
    const float* Wfc_p = (const float*)d_in[5];
    const float* bfc_p = (const float*)d_in[6];

    const int B = out_size;              // O == 1
    const int T = in_sizes[0] / B;       // I == 1  -> T = 512

    dim3 grid(B / 16), block(32);
    WeatherLSTM_wmma_kernel<<<grid, block, 0, stream>>>(
        x_p, Wih_p, Whh_p, bih_p, bhh_p, Wfc_p, bfc_p, (float*)d_out, T);
}